// TransitionProbLayerBeta_33672543600698
// MI455X (gfx1250) — compile-verified
//
#include <hip/hip_runtime.h>
#include <hip/hip_bf16.h>

typedef __attribute__((ext_vector_type(2))) float v2f;
typedef __attribute__((ext_vector_type(8))) float v8f;

#define N_CAMP 50
#define DAYS   365
#define DP1    366
#define BATCH  4096
#define ROWS   (BATCH * DAYS)               /* 1,495,040 (b,d) rows          */
#define WAVES_PER_BLOCK 8
#define TILES_PER_WAVE  4
#define TILES  (ROWS / 16)                  /* 93,440 — divides exactly      */
#define NBLOCKS (TILES / (WAVES_PER_BLOCK * TILES_PER_WAVE))   /* 2,920     */
#define BASIS  1e-10f
#define QN     ((size_t)ROWS * 9)           /* 13,455,360 Q elements         */
#define ZSLOT  (N_CAMP * 3)                 /* LDS zero slot index           */

// One wave32 processes 4 tiles of 16 (b,d)-rows.  Per tile: issue all 26
// adstock loads as one clause (max MLP), drain through 13 chained
// v_wmma_f32_16x16x4_f32 (K = 50 campaigns padded to 52), then the 3x3
// transition-matrix epilogue.  B fragments (betas) are hoisted to registers
// once per wave via an LDS zero-slot so no exec-mask divergence remains in
// the hot path.
__global__ __launch_bounds__(256) void transition_wmma_kernel(
    const float* __restrict__ adstock,     // [4096, 50, 366]
    const float* __restrict__ mu,          // [3]
    const float* __restrict__ beta,        // [150] = [50,3] row-major
    const float* __restrict__ init_prob,   // [1]
    const float* __restrict__ click_prob,  // [1]
    float* __restrict__ out)               // [ROWS*9] Q ++ [2] sigmoids
{
    __shared__ float s_beta[N_CAMP * 3 + 1];      // +1 zero slot
    __shared__ float s_O[WAVES_PER_BLOCK][48];    // 16 rows x {O1,O2,O3}
    __shared__ float s_Q[WAVES_PER_BLOCK][144];   // 16 rows x 9 (staging)

    const int tid  = threadIdx.x;
    const int wave = tid >> 5;
    const int lane = tid & 31;

    if (tid < N_CAMP * 3) s_beta[tid] = beta[tid];
    if (tid == ZSLOT)     s_beta[ZSLOT] = 0.0f;
    __syncthreads();

    const int m     = lane & 15;           // M row (A) == N col (B/C/D)
    const int half  = lane >> 4;           // A/B: K pair {0,1} vs {2,3}
    const int cHalf = half * 2;

    // --- Hoist all B fragments (loop/tile-invariant per lane) to registers.
    // B 4x16 assumed to mirror the A 16x4 layout: VGPR0 = row K{0|2},
    // VGPR1 = row K{1|3}, N = lane%16.  beta_r(c,n) = beta[c*3+n] for n<3,
    // zero column elsewhere -> zero-slot select, no branches.
    float bxr[13], byr[13];
#pragma unroll
    for (int kk = 0; kk < 13; ++kk) {
        const int c0 = kk * 4 + cHalf;
        const int i0 = (m < 3 && c0     < N_CAMP) ? c0 * 3 + m       : ZSLOT;
        const int i1 = (m < 3 && c0 + 1 < N_CAMP) ? (c0 + 1) * 3 + m : ZSLOT;
        bxr[kk] = s_beta[i0];
        byr[kk] = s_beta[i1];
    }

    const float mu0 = mu[0], mu1 = mu[1], mu2 = mu[2];
    const int tile0 = (blockIdx.x * WAVES_PER_BLOCK + wave) * TILES_PER_WAVE;

#pragma unroll 1
    for (int j = 0; j < TILES_PER_WAVE; ++j) {
        const int tile = tile0 + j;
        const int r = tile * 16 + m;       // global (b,d) row for A fragment
        const int b = r / DAYS;
        const int d = r - b * DAYS;
        const float* base = adstock + (size_t)b * (N_CAMP * DP1) + (size_t)(d + 1);
        const float* p    = base + cHalf * DP1;          // per-half K base
        const float* pend = base + (N_CAMP - 2) * DP1;   // c=48: kk=12 clamp

        // --- One load clause: 26 coalesced dword loads (contiguous along d
        // across each 16-lane half).  kk=12/half1 would touch campaigns
        // 50/51; their B rows are the zero slot, so clamp the address to
        // valid rows 48/49 (finite don't-cares) instead of masking.
        float a0[13], a1[13];
#pragma unroll
        for (int kk = 0; kk < 12; ++kk) {
            a0[kk] = p[(size_t)(kk * 4)     * DP1];
            a1[kk] = p[(size_t)(kk * 4 + 1) * DP1];
        }
        a0[12] = pend[0];                  // half0: c=48 real; half1: B=0
        a1[12] = pend[DP1];                // half0: c=49 real; half1: B=0

        // --- 13 chained f32 WMMAs drain the loads.
        v8f acc = {};
#pragma unroll
        for (int kk = 0; kk < 13; ++kk) {
            v2f a  = {a0[kk],  a1[kk]};
            v2f bb = {bxr[kk], byr[kk]};
            acc = __builtin_amdgcn_wmma_f32_16x16x4_f32(
                /*neg_a=*/false, a, /*neg_b=*/false, bb,
                /*c_mod=*/(short)0, acc, /*reuse_a=*/false, /*reuse_b=*/false);
        }

        // --- Gather the 3 useful accumulator columns (N=0..2) into LDS.
        // C/D layout: VGPR i -> M = i + 8*half, N = lane%16.
        if (m < 3) {
#pragma unroll
            for (int i = 0; i < 8; ++i)
                s_O[wave][(half * 8 + i) * 3 + m] = acc[i];
        }
        __syncthreads();

        // --- One lane per row builds the 3x3 transition matrix.
        if (lane < 16) {
            const float O1 = s_O[wave][lane * 3 + 0];
            const float O2 = s_O[wave][lane * 3 + 1];
            const float O3 = s_O[wave][lane * 3 + 2];
            const float e1 = expf(mu0 + O1);
            const float e2 = expf(mu1 + O2);
            const float e3 = expf(mu2 + O3);
            const float i0 = 1.0f / (1.0f + e1);       // den0 (exp(-inf)=0 slot)
            const float i1 = 1.0f / (1.0f + e2 + e3);  // den1
            float q[9];
            q[0] = fmaxf(i0,      BASIS);  q[1] = fmaxf(e1 * i0, BASIS);  q[2] = BASIS;
            q[3] = fmaxf(e2 * i1, BASIS);  q[4] = fmaxf(i1,      BASIS);  q[5] = fmaxf(e3 * i1, BASIS);
            q[6] = BASIS;                  q[7] = BASIS;                  q[8] = 1.0f;
#pragma unroll
            for (int jj = 0; jj < 9; ++jj) s_Q[wave][lane * 9 + jj] = q[jj];
        }
        __syncthreads();

        // --- Coalesced tile store: 144 contiguous floats, all 32 lanes.
        const size_t ob = (size_t)tile * 144;
#pragma unroll
        for (int idx = lane; idx < 144; idx += 32)
            out[ob + idx] = s_Q[wave][idx];
    }

    // --- Scalar sigmoids appended after Q (reference return order).
    if (blockIdx.x == 0 && tid == 0) {
        out[QN]     = 1.0f / (1.0f + expf(-init_prob[0]));
        out[QN + 1] = 1.0f / (1.0f + expf(-click_prob[0]));
    }
}

extern "C" void kernel_launch(void* const* d_in, const int* in_sizes, int n_in,
                              void* d_out, int out_size, void* d_ws, size_t ws_size,
                              hipStream_t stream) {
    (void)in_sizes; (void)n_in; (void)out_size; (void)d_ws; (void)ws_size;
    const float* adstock    = (const float*)d_in[0];
    const float* mu         = (const float*)d_in[1];
    const float* beta       = (const float*)d_in[2];
    const float* init_prob  = (const float*)d_in[3];
    const float* click_prob = (const float*)d_in[4];
    float* out = (float*)d_out;
    transition_wmma_kernel<<<NBLOCKS, 256, 0, stream>>>(
        adstock, mu, beta, init_prob, click_prob, out);
}